// AtomicHAR_39668317946472
// MI455X (gfx1250) — compile-verified
//
#include <hip/hip_runtime.h>
#include <math.h>

typedef float v2f __attribute__((ext_vector_type(2)));
typedef float v8f __attribute__((ext_vector_type(8)));

// D = A(16x4 f32) * B(4x16 f32) + C(16x16 f32)
#define WMMA_F32(a, b, c) \
  __builtin_amdgcn_wmma_f32_16x16x4_f32(false, (a), false, (b), (short)0, (c), false, false)

// ---------------- problem constants ----------------
constexpr int BS = 32, SEQ = 256, DIMC = 6, L = 400;
constexpr int DM = 4, DFF = 16, DOUT = 32;
constexpr int MAXA = SEQ / 2 + 2;   // 130
constexpr int ILEN = 20;
constexpr float THR = 0.001f;
constexpr int N = BS * SEQ;         // 8192
constexpr int OC = 32, KW = 3, OT = L - KW + 1; // 398
constexpr int HALF = OT / 2;        // 199
constexpr int PADL = L + 4;         // 404 (covers t+k overrun with zeros)
constexpr int DO2 = DIMC * L;       // 2400

// ---------------- workspace layout (float units) ----------------
constexpr size_t WS_BRIDGE = 0;                             // N*4 f32
constexpr size_t WS_H      = WS_BRIDGE + (size_t)N * 4;     // N*64 f32
constexpr size_t WS_TROUT  = WS_H + (size_t)N * 64;         // N*32 f32
constexpr size_t WS_SEGID  = WS_TROUT + (size_t)N * 32;     // N i32
constexpr size_t WS_LAST   = WS_SEGID + N;                  // BS i32
constexpr size_t WS_NK     = WS_LAST + BS;                  // BS i32
constexpr size_t WS_ENDS   = WS_NK + BS;                    // BS*MAXA i32
constexpr size_t WS_STARTS = WS_ENDS + (size_t)BS * MAXA;   // BS*MAXA i32

// ---------------- output layout (float units) ----------------
constexpr size_t O_IMU    = 0;
constexpr size_t O_ATOM   = O_IMU + (size_t)N * DO2;                 // 19,660,800
constexpr size_t O_INTERP = O_ATOM + (size_t)BS * MAXA * DIMC * ILEN;
constexpr size_t O_FC     = O_INTERP + (size_t)BS * MAXA * DIMC * ILEN;
constexpr size_t O_FLOSS  = O_FC + (size_t)N * DM;

// =====================================================================
// Kernel 1: conv1d(+bias,relu) -> half-pool -> sigmoid bridge -> hidden
// One wave per sample; implicit-GEMM conv with f32 WMMA (K padded 18->20).
// =====================================================================
__global__ __launch_bounds__(128) void k_conv_bridge(
    const float* __restrict__ x, const float* __restrict__ conv_w,
    const float* __restrict__ conv_b, const float* __restrict__ W_b1,
    const float* __restrict__ b_b1, const float* __restrict__ Wd1,
    const float* __restrict__ bd1, float* __restrict__ ws) {
  __shared__ float xs[4 * DIMC * PADL];
  const int lane = threadIdx.x & 31;
  const int wv = threadIdx.x >> 5;
  const int n = blockIdx.x * 4 + wv;
  float* xw = xs + wv * DIMC * PADL;

  // stage sample (6 x 400) into LDS, rows padded to 404 with zeros
  const float* xn = x + (size_t)n * (DIMC * L);
  for (int d = 0; d < DIMC; ++d) {
    const float4* src = (const float4*)(xn + d * L);
    float4* dst = (float4*)(xw + d * PADL);
    for (int i = lane; i < L / 4; i += 32) dst[i] = src[i];
    if (lane < 4) xw[d * PADL + L + lane] = 0.f;
  }
  __syncthreads();

  const int cl = lane & 15;
  const int kh = (lane >> 4) * 2;  // K half offset for A/B fragments

  // B fragments: B[kk][c] = conv_w[c][kk], kk in [0,20) padded
  v2f bf[2][5];
#pragma unroll
  for (int nt = 0; nt < 2; ++nt) {
    int c = nt * 16 + cl;
#pragma unroll
    for (int ks = 0; ks < 5; ++ks) {
      int k0 = ks * 4 + kh;
      float bx = (k0 < 18) ? conv_w[c * 18 + k0] : 0.f;
      float by = (k0 + 1 < 18) ? conv_w[c * 18 + k0 + 1] : 0.f;
      v2f t = {bx, by};
      bf[nt][ks] = t;
    }
  }
  const float cb0 = conv_b[cl], cb1 = conv_b[16 + cl];

  // A fragment LDS offsets per K-step (kk -> (d = kk/3, k = kk%3));
  // invalid (padded) kk fall back to in-bounds offset 0 and are zeroed by
  // an unconditional select (no exec-mask branches in the hot loop).
  int aoff0[5], aoff1[5];
  bool av0[5], av1[5];
#pragma unroll
  for (int ks = 0; ks < 5; ++ks) {
    int k0 = ks * 4 + kh;
    av0[ks] = (k0 < 18);
    av1[ks] = (k0 + 1 < 18);
    int kk0 = av0[ks] ? k0 : 0, kk1 = av1[ks] ? (k0 + 1) : 0;
    aoff0[ks] = (kk0 / 3) * PADL + (kk0 % 3);
    aoff1[ks] = (kk1 / 3) * PADL + (kk1 % 3);
  }

  const int m = cl;
  float hs0[2] = {0.f, 0.f};  // channel cl     (tile 0) half sums
  float hs1[2] = {0.f, 0.f};  // channel 16+cl  (tile 1) half sums

  for (int t0 = 0; t0 < OT; t0 += 16) {  // 25 M-tiles
    v8f acc0 = {}, acc1 = {};
#pragma unroll
    for (int ks = 0; ks < 5; ++ks) {
      float lx = xw[aoff0[ks] + t0 + m];   // unconditional, always in-bounds
      float ly = xw[aoff1[ks] + t0 + m];
      float ax = av0[ks] ? lx : 0.f;
      float ay = av1[ks] ? ly : 0.f;
      v2f a = {ax, ay};
      acc0 = WMMA_F32(a, bf[0][ks], acc0);
      acc1 = WMMA_F32(a, bf[1][ks], acc1);
    }
    int hi = (lane >> 4) * 8;
#pragma unroll
    for (int r = 0; r < 8; ++r) {
      int t = t0 + r + hi;
      if (t < OT) {
        float y0 = fmaxf(acc0[r] + cb0, 0.f);
        float y1 = fmaxf(acc1[r] + cb1, 0.f);
        int h = (t < HALF) ? 0 : 1;
        hs0[h] += y0;
        hs1[h] += y1;
      }
    }
  }
  // combine the two K-half lanes holding the same channel
  hs0[0] += __shfl_xor(hs0[0], 16);
  hs0[1] += __shfl_xor(hs0[1], 16);
  hs1[0] += __shfl_xor(hs1[0], 16);
  hs1[1] += __shfl_xor(hs1[1], 16);

  // lane L owns channel L
  float p0 = (lane < 16) ? hs0[0] : hs1[0];
  float p1 = (lane < 16) ? hs0[1] : hs1[1];
  const float inv = 1.f / (float)HALF;
  p0 *= inv;
  p1 *= inv;

  // bridge: sigmoid(cnn_flat @ W_b1 + b_b1), cnn_flat[2c]=p0, [2c+1]=p1
  float part[4];
#pragma unroll
  for (int j = 0; j < 4; ++j)
    part[j] = p0 * W_b1[(2 * lane) * 4 + j] + p1 * W_b1[(2 * lane + 1) * 4 + j];
#pragma unroll
  for (int off = 16; off >= 1; off >>= 1) {
#pragma unroll
    for (int j = 0; j < 4; ++j) part[j] += __shfl_xor(part[j], off);
  }
  float br[4];
#pragma unroll
  for (int j = 0; j < 4; ++j) br[j] = 1.f / (1.f + expf(-(part[j] + b_b1[j])));
  if (lane == 0) {
    float4 bv = {br[0], br[1], br[2], br[3]};
    *(float4*)(ws + WS_BRIDGE + (size_t)n * 4) = bv;
  }
  // decoder hidden: relu(bridge @ Wd1 + bd1), 64 wide
  float* hws = ws + WS_H + (size_t)n * 64;
#pragma unroll
  for (int rep = 0; rep < 2; ++rep) {
    int i = lane + rep * 32;
    float acc = bd1[i];
#pragma unroll
    for (int j = 0; j < 4; ++j) acc += br[j] * Wd1[j * 64 + i];
    hws[i] = fmaxf(acc, 0.f);
  }
}

// =====================================================================
// Kernel 2: forcast + masked/thresholded floss (written to out regions)
// =====================================================================
__global__ __launch_bounds__(256) void k_forcast(
    const float* __restrict__ ws, const float* __restrict__ imu_mask,
    const float* __restrict__ W_fc, const float* __restrict__ b_fc,
    float* __restrict__ out) {
  int n = blockIdx.x * blockDim.x + threadIdx.x;
  if (n >= N) return;
  int s = n & (SEQ - 1);
  const float* cur = ws + WS_BRIDGE + (size_t)n * 4;
  // previous-step bridge, kept in-bounds (masked by smask, no speculation OOB)
  const float smask = (s == 0) ? 0.f : 1.f;
  const float* prev = ws + WS_BRIDGE + (size_t)((s == 0) ? n : (n - 1)) * 4;
  float sh[4];
#pragma unroll
  for (int j = 0; j < 4; ++j) sh[j] = prev[j] * smask;
  float fc[4];
#pragma unroll
  for (int j = 0; j < 4; ++j) {
    float a = b_fc[j];
#pragma unroll
    for (int d = 0; d < 4; ++d) a += sh[d] * W_fc[d * 4 + j];
    fc[j] = a;
    out[O_FC + (size_t)n * 4 + j] = a;
  }
  float fm = smask * imu_mask[n];
  float fl = 0.f;
#pragma unroll
  for (int j = 0; j < 4; ++j) {
    float d = (fc[j] - cur[j]) * fm;
    fl += d * d;
  }
  fl *= 0.25f;
  bool lm = (s >= 2) && (s <= SEQ - 3);
  out[O_FLOSS + n] = (fl > THR && lm) ? fl : 0.f;
}

// =====================================================================
// Kernel 3: segmentation (per-batch sequential, lane b = batch b)
// =====================================================================
__global__ void k_seg(const float* __restrict__ out, const int* __restrict__ imu_len,
                      float* __restrict__ ws) {
  int b = threadIdx.x;
  if (b >= BS) return;
  int* segid = (int*)(ws + WS_SEGID);
  int* lastp = (int*)(ws + WS_LAST);
  int* nkp = (int*)(ws + WS_NK);
  int* ends = (int*)(ws + WS_ENDS);
  int* starts = (int*)(ws + WS_STARTS);
  const float* fl = out + O_FLOSS + (size_t)b * SEQ;

  float sel[SEQ];
  for (int w = 0; w < SEQ / 4; ++w) {
    int base = w * 4;
    int mi = 0;
    float mv = fl[base];
    for (int i = 1; i < 4; ++i)
      if (fl[base + i] > mv) { mv = fl[base + i]; mi = i; }
    for (int i = 0; i < 4; ++i) sel[base + i] = (i == mi) ? fl[base + i] : 0.f;
  }
  bool segpt[SEQ];
  for (int s = 0; s < SEQ; ++s) segpt[s] = false;
  for (int w = 0; w < 63; ++w) {
    int base = 2 + w * 4;
    int mi = 0;
    float mv = sel[base];
    for (int i = 1; i < 4; ++i)
      if (sel[base + i] > mv) { mv = sel[base + i]; mi = i; }
    if (sel[base + mi] > 0.f) segpt[base + mi] = true;
  }
  int last = (int)rintf((float)imu_len[b] / (float)SEQ);  // round half-to-even
  last = min(max(last, 2), SEQ);

  bool point[SEQ];
  for (int s = 0; s < SEQ; ++s) point[s] = segpt[s] && (s < last);

  int nk = 0, sid = 0;
  int kp[MAXA];
  for (int s = 0; s < SEQ; ++s) {
    bool bn = ((s + 1 < SEQ) ? point[s + 1] : false) || (s + 1 == last);
    bool k = point[s] && !bn;
    if (k) {
      ++sid;
      if (nk < MAXA) kp[nk++] = s;
    }
    segid[b * SEQ + s] = sid;
  }
  lastp[b] = last;
  nkp[b] = nk;
  for (int a = 0; a < MAXA; ++a) {
    int e = (a < nk) ? kp[a] : last;
    ends[b * MAXA + a] = e;
    starts[b * MAXA + a] = (a == 0) ? 0 : ends[b * MAXA + a - 1];
  }
}

// =====================================================================
// Kernel 4: tiny transformer (block = batch; thread = query position)
// NOTE reference reads bridge via reshape(seq, bs, DM): hb[b][s] = bridge[s*BS+b]
// =====================================================================
__device__ __forceinline__ void ln4(const float* xin, const float* g,
                                    const float* bb, float* yo) {
  float m = 0.25f * (xin[0] + xin[1] + xin[2] + xin[3]);
  float v = 0.f;
#pragma unroll
  for (int d = 0; d < 4; ++d) {
    float dd = xin[d] - m;
    v += dd * dd;
  }
  v *= 0.25f;
  float r = rsqrtf(v + 1e-5f);
#pragma unroll
  for (int d = 0; d < 4; ++d) yo[d] = (xin[d] - m) * r * g[d] + bb[d];
}

__global__ __launch_bounds__(256) void k_xform(
    const float* __restrict__ Wqkv, const float* __restrict__ Wo,
    const float* __restrict__ ln1g, const float* __restrict__ ln1b,
    const float* __restrict__ Wf1, const float* __restrict__ bf1,
    const float* __restrict__ Wf2, const float* __restrict__ bf2,
    const float* __restrict__ ln2g, const float* __restrict__ ln2b,
    const float* __restrict__ Wout, const float* __restrict__ bout,
    float* __restrict__ ws) {
  __shared__ float hb[SEQ][DM], kk[SEQ][DM], vv[SEQ][DM];
  __shared__ int sid[SEQ];
  __shared__ int last_s;
  const int b = blockIdx.x, s = threadIdx.x;

  const float* brp = ws + WS_BRIDGE + ((size_t)s * BS + b) * 4;
  float h[4], q[4];
#pragma unroll
  for (int d = 0; d < 4; ++d) h[d] = brp[d];
#pragma unroll
  for (int e = 0; e < 4; ++e) {
    float aq = 0.f, ak = 0.f, av = 0.f;
#pragma unroll
    for (int d = 0; d < 4; ++d) {
      aq += h[d] * Wqkv[0 * 16 + d * 4 + e];
      ak += h[d] * Wqkv[1 * 16 + d * 4 + e];
      av += h[d] * Wqkv[2 * 16 + d * 4 + e];
    }
    q[e] = aq;
    kk[s][e] = ak;
    vv[s][e] = av;
  }
#pragma unroll
  for (int d = 0; d < 4; ++d) hb[s][d] = h[d];
  sid[s] = ((const int*)(ws + WS_SEGID))[b * SEQ + s];
  if (s == 0) last_s = ((const int*)(ws + WS_LAST))[b];
  __syncthreads();

  const int lastb = last_s;
  const bool vq = (s < lastb);
  const int ms = sid[s];
  const float scale = 0.70710678118654752440f;  // 1/sqrt(hd), hd=2
  float ao[4];
#pragma unroll
  for (int hh = 0; hh < 2; ++hh) {
    float q0 = q[hh * 2], q1 = q[hh * 2 + 1];
    float mx = -INFINITY;
    for (int j = 0; j < SEQ; ++j) {
      bool al = (j == s) || (vq && (j < lastb) && (sid[j] == ms));
      if (al) {
        float sc = (q0 * kk[j][hh * 2] + q1 * kk[j][hh * 2 + 1]) * scale;
        mx = fmaxf(mx, sc);
      }
    }
    float sm = 0.f, o0 = 0.f, o1 = 0.f;
    for (int j = 0; j < SEQ; ++j) {
      bool al = (j == s) || (vq && (j < lastb) && (sid[j] == ms));
      if (al) {
        float sc = (q0 * kk[j][hh * 2] + q1 * kk[j][hh * 2 + 1]) * scale;
        float e = expf(sc - mx);
        sm += e;
        o0 += e * vv[j][hh * 2];
        o1 += e * vv[j][hh * 2 + 1];
      }
    }
    ao[hh * 2] = o0 / sm;
    ao[hh * 2 + 1] = o1 / sm;
  }
  float x1[4], h1[4];
#pragma unroll
  for (int d = 0; d < 4; ++d) {
    float a = hb[s][d];
#pragma unroll
    for (int t = 0; t < 4; ++t) a += ao[t] * Wo[t * 4 + d];
    x1[d] = a;
  }
  ln4(x1, ln1g, ln1b, h1);
  float f[DFF];
#pragma unroll
  for (int u = 0; u < DFF; ++u) {
    float a = bf1[u];
#pragma unroll
    for (int d = 0; d < 4; ++d) a += h1[d] * Wf1[d * DFF + u];
    f[u] = fmaxf(a, 0.f);
  }
  float x2[4], h2[4];
#pragma unroll
  for (int d = 0; d < 4; ++d) {
    float a = h1[d] + bf2[d];
#pragma unroll
    for (int u = 0; u < DFF; ++u) a += f[u] * Wf2[u * 4 + d];
    x2[d] = a;
  }
  ln4(x2, ln2g, ln2b, h2);
  float* tro = ws + WS_TROUT + ((size_t)(b * SEQ + s)) * DOUT;
#pragma unroll
  for (int o = 0; o < DOUT; ++o) {
    float a = bout[o];
#pragma unroll
    for (int d = 0; d < 4; ++d) a += h2[d] * Wout[d * DOUT + o];
    tro[o] = a;
  }
}

// =====================================================================
// Kernel 5: atom_gen = (emb @ Wa + ba) * atom_valid
// =====================================================================
__global__ __launch_bounds__(256) void k_atom(const float* __restrict__ ws,
                                              const float* __restrict__ Wa,
                                              const float* __restrict__ ba,
                                              float* __restrict__ out) {
  int tid = blockIdx.x * blockDim.x + threadIdx.x;  // exactly BS*MAXA*120
  int j = tid % 120;
  int a = (tid / 120) % MAXA;
  int b = tid / (120 * MAXA);
  const int* ends = (const int*)(ws + WS_ENDS);
  const int* nkp = (const int*)(ws + WS_NK);
  int e = ends[b * MAXA + a];
  int row = min(max(e - 1, 0), SEQ - 1);
  float valid = (a <= nkp[b]) ? 1.f : 0.f;
  const float* tr = ws + WS_TROUT + ((size_t)(b * SEQ + row)) * DOUT;
  float acc = ba[j];
#pragma unroll
  for (int t = 0; t < DOUT; ++t) acc += tr[t] * Wa[t * 120 + j];
  out[O_ATOM + (size_t)(b * MAXA + a) * 120 + j] = acc * valid;
}

// =====================================================================
// Kernel 6: seg_interp gather
// =====================================================================
__global__ __launch_bounds__(256) void k_interp(const float* __restrict__ x,
                                                const float* __restrict__ ws,
                                                float* __restrict__ out) {
  int tid = blockIdx.x * blockDim.x + threadIdx.x;  // BS*MAXA*DIMC*ILEN
  int i = tid % ILEN;
  int d = (tid / ILEN) % DIMC;
  int a = (tid / (ILEN * DIMC)) % MAXA;
  int b = tid / (ILEN * DIMC * MAXA);
  const int* ends = (const int*)(ws + WS_ENDS);
  const int* starts = (const int*)(ws + WS_STARTS);
  const int* nkp = (const int*)(ws + WS_NK);
  int st = starts[b * MAXA + a], en = ends[b * MAXA + a];
  int inl = (en - st) * L;
  int idx = st * L + (i * inl) / ILEN;
  idx = min(max(idx, 0), SEQ * L - 1);
  float valid = (a <= nkp[b]) ? 1.f : 0.f;
  int srow = idx / L, scol = idx % L;
  float v = x[(((size_t)b * SEQ + srow) * DIMC + d) * L + scol];
  out[O_INTERP + ((size_t)(b * MAXA + a) * DIMC + d) * ILEN + i] = v * valid;
}

// =====================================================================
// Kernel 7: imu_gen GEMM via f32 WMMA: (8192 x 64) @ (64 x 2400) + bd2
// Block = 8 waves sharing one 32-col Wd2 panel staged in LDS;
// wave = 16-row x 32-col C strip; 16 K-steps x 2 tiles = 32 WMMAs.
// =====================================================================
__global__ __launch_bounds__(256) void k_imugen(const float* __restrict__ ws,
                                                const float* __restrict__ Wd2,
                                                const float* __restrict__ bd2,
                                                float* __restrict__ out) {
  __shared__ float bpan[64 * 32];  // 8 KB: Wd2[k][tp*32 .. +32)
  const int tp = blockIdx.x % 75;       // column panel (32 cols)
  const int tmBase = (blockIdx.x / 75) * 8;  // 8 row tiles per block
  {
    int c = threadIdx.x & 31;
    int r0 = threadIdx.x >> 5;
    const float* src = Wd2 + (size_t)tp * 32 + c;
    for (int r = r0; r < 64; r += 8) bpan[r * 32 + c] = src[(size_t)r * DO2];
  }
  __syncthreads();

  const int lane = threadIdx.x & 31;
  const int tm = tmBase + (threadIdx.x >> 5);
  const int cl = lane & 15;
  const int kh = (lane >> 4) * 2;
  const int row_a = tm * 16 + cl;
  const int col0 = tp * 32 + cl, col1 = col0 + 16;
  const float* ha = ws + WS_H + (size_t)row_a * 64;
  __builtin_prefetch(ha, 0, 0);  // global_prefetch_b8 of this wave's A tile

  v8f acc0 = {}, acc1 = {};
#pragma unroll
  for (int ks = 0; ks < 16; ++ks) {
    int k0 = ks * 4 + kh;
    v2f a = *(const v2f*)(ha + k0);
    v2f b0 = {bpan[k0 * 32 + cl], bpan[(k0 + 1) * 32 + cl]};
    v2f b1 = {bpan[k0 * 32 + 16 + cl], bpan[(k0 + 1) * 32 + 16 + cl]};
    acc0 = WMMA_F32(a, b0, acc0);
    acc1 = WMMA_F32(a, b1, acc1);
  }
  const float bb0 = bd2[col0], bb1 = bd2[col1];
  const int hi = (lane >> 4) * 8;
#pragma unroll
  for (int r = 0; r < 8; ++r) {
    int row = tm * 16 + r + hi;
    float* po = out + O_IMU + (size_t)row * DO2;
    po[col0] = acc0[r] + bb0;
    po[col1] = acc1[r] + bb1;
  }
}

// =====================================================================
extern "C" void kernel_launch(void* const* d_in, const int* in_sizes, int n_in,
                              void* d_out, int out_size, void* d_ws, size_t ws_size,
                              hipStream_t stream) {
  const float* x = (const float*)d_in[0];
  const float* imu_mask = (const float*)d_in[1];
  const int* imu_len = (const int*)d_in[2];
  const float* conv_w = (const float*)d_in[3];
  const float* conv_b = (const float*)d_in[4];
  const float* W_b1 = (const float*)d_in[5];
  const float* b_b1 = (const float*)d_in[6];
  const float* W_fc = (const float*)d_in[7];
  const float* b_fc = (const float*)d_in[8];
  const float* Wqkv = (const float*)d_in[9];
  const float* Wo = (const float*)d_in[10];
  const float* ln1_g = (const float*)d_in[11];
  const float* ln1_b = (const float*)d_in[12];
  const float* Wf1 = (const float*)d_in[13];
  const float* bf1 = (const float*)d_in[14];
  const float* Wf2 = (const float*)d_in[15];
  const float* bf2 = (const float*)d_in[16];
  const float* ln2_g = (const float*)d_in[17];
  const float* ln2_b = (const float*)d_in[18];
  const float* Wout = (const float*)d_in[19];
  const float* bout = (const float*)d_in[20];
  const float* Wd1 = (const float*)d_in[21];
  const float* bd1 = (const float*)d_in[22];
  const float* Wd2 = (const float*)d_in[23];
  const float* bd2 = (const float*)d_in[24];
  const float* Wa = (const float*)d_in[25];
  const float* ba = (const float*)d_in[26];

  float* outf = (float*)d_out;
  float* wsf = (float*)d_ws;

  // 1) conv + bridge + decoder hidden (WMMA f32 implicit GEMM)
  k_conv_bridge<<<N / 4, 128, 0, stream>>>(x, conv_w, conv_b, W_b1, b_b1, Wd1,
                                           bd1, wsf);
  // 2) forcast + floss
  k_forcast<<<N / 256, 256, 0, stream>>>(wsf, imu_mask, W_fc, b_fc, outf);
  // 3) segmentation
  k_seg<<<1, 32, 0, stream>>>(outf, imu_len, wsf);
  // 4) transformer -> tr_out
  k_xform<<<BS, SEQ, 0, stream>>>(Wqkv, Wo, ln1_g, ln1_b, Wf1, bf1, Wf2, bf2,
                                  ln2_g, ln2_b, Wout, bout, wsf);
  // 5) atom_gen
  k_atom<<<(BS * MAXA * 120) / 256, 256, 0, stream>>>(wsf, Wa, ba, outf);
  // 6) seg_interp
  k_interp<<<(BS * MAXA * DIMC * ILEN) / 256, 256, 0, stream>>>(x, wsf, outf);
  // 7) imu_gen (WMMA f32 GEMM, 78.6 MB store-bound)
  k_imugen<<<(512 * 75) / 8, 256, 0, stream>>>(wsf, Wd2, bd2, outf);
}